// LAEConvOperation_85787676770352
// MI455X (gfx1250) — compile-verified
//
#include <hip/hip_runtime.h>
#include <math.h>

typedef float v2f __attribute__((ext_vector_type(2)));
typedef float v8f __attribute__((ext_vector_type(8)));

// Problem constants (from setup_inputs)
constexpr int B   = 32;
constexpr int C   = 512;
constexpr int N   = 8192;
constexpr int O   = 512;
constexpr int KNN = 256;

// Workspace layout (in floats)
constexpr size_t OFF_D2  = 0;                         // B*N
constexpr size_t OFF_IDX = OFF_D2  + (size_t)B * N;   // B*KNN (ints)
constexpr size_t OFF_REL = OFF_IDX + (size_t)B * KNN; // B*C*KNN
constexpr size_t OFF_T   = OFF_REL + (size_t)B * C * KNN;   // B*O*KNN
constexpr size_t OFF_S   = OFF_T   + (size_t)B * O * KNN;   // B*KNN*KNN
constexpr size_t OFF_W   = OFF_S   + (size_t)B * KNN * KNN; // B*KNN

// ---------------- wave helpers (wave32) ----------------
__device__ __forceinline__ float wave_sum(float v) {
#pragma unroll
  for (int o = 16; o > 0; o >>= 1) v += __shfl_xor(v, o, 32);
  return v;
}
__device__ __forceinline__ float wave_max(float v) {
#pragma unroll
  for (int o = 16; o > 0; o >>= 1) v = fmaxf(v, __shfl_xor(v, o, 32));
  return v;
}

// ---------------- 1) squared distances d2[b][n], float4-vectorized stream ----------------
__global__ __launch_bounds__(256) void dist2_kernel(const float* __restrict__ h,
                                                    const float* __restrict__ pi,
                                                    float* __restrict__ d2) {
  __shared__ float spi[C];
  const int b = blockIdx.x >> 3;          // 8 n-chunks of 1024 per batch
  const int chunk = blockIdx.x & 7;
  const int tid = threadIdx.x;
  spi[tid]       = pi[b * C + tid];
  spi[tid + 256] = pi[b * C + tid + 256];
  __syncthreads();
  const int n0 = chunk * 1024 + tid * 4;  // 4 consecutive points per thread
  const float4* hb = (const float4*)(h + (size_t)b * C * N + n0);
  float a0 = 0.f, a1 = 0.f, a2 = 0.f, a3 = 0.f;
#pragma unroll 4
  for (int c = 0; c < C; ++c) {
    float4 v = hb[(size_t)c * (N / 4)];   // global_load_b128, 512B per wave
    const float p = spi[c];
    float d0 = v.x - p, d1 = v.y - p, d2v = v.z - p, d3 = v.w - p;
    a0 = fmaf(d0, d0, a0); a1 = fmaf(d1, d1, a1);
    a2 = fmaf(d2v, d2v, a2); a3 = fmaf(d3, d3, a3);
  }
  float4 r; r.x = a0; r.y = a1; r.z = a2; r.w = a3;
  *(float4*)(d2 + b * N + n0) = r;
}

// ---------------- 2) per-batch top-K (order-free set) via LDS radix select ----------------
__global__ __launch_bounds__(1024) void topk_kernel(const float* __restrict__ d2,
                                                    int* __restrict__ idx) {
  __shared__ unsigned skey[N];            // 32 KB of the 320 KB LDS
  __shared__ unsigned hist[256];
  __shared__ unsigned sPrefix, sMask, cLess, cEq;
  __shared__ int sRemK;
  const int b = blockIdx.x;
  const int tid = threadIdx.x;
  for (int n = tid; n < N; n += 1024)
    skey[n] = __float_as_uint(d2[b * N + n]);    // d2 >= 0 -> bit order == value order
  if (tid == 0) { sPrefix = 0u; sMask = 0u; sRemK = KNN; }
  __syncthreads();
  for (int pass = 0; pass < 4; ++pass) {
    const int shift = 24 - 8 * pass;
    if (tid < 256) hist[tid] = 0u;
    __syncthreads();
    const unsigned pfx = sPrefix, msk = sMask;
    for (int n = tid; n < N; n += 1024) {
      unsigned k = skey[n];
      if ((k & msk) == pfx) atomicAdd(&hist[(k >> shift) & 255u], 1u);
    }
    __syncthreads();
    if (tid == 0) {
      int rem = sRemK; unsigned run = 0; int d = 0;
      for (; d < 256; ++d) {
        unsigned hc = hist[d];
        if (run + hc >= (unsigned)rem) { sRemK = rem - (int)run; break; }
        run += hc;
      }
      sPrefix = pfx | ((unsigned)d << shift);
      sMask   = msk | (255u << shift);
    }
    __syncthreads();
  }
  const unsigned thr = sPrefix;           // exact K-th smallest key
  const int numEq = sRemK, numLess = KNN - numEq;
  if (tid == 0) { cLess = 0u; cEq = 0u; }
  __syncthreads();
  for (int n = tid; n < N; n += 1024) {
    unsigned k = skey[n];
    if (k < thr) {
      unsigned p = atomicAdd(&cLess, 1u);
      idx[b * KNN + p] = n;
    } else if (k == thr) {
      unsigned e = atomicAdd(&cEq, 1u);
      if (e < (unsigned)numEq) idx[b * KNN + numLess + e] = n;
    }
  }
}

// ---------------- 3) packed gather: rel[b][c][k] = h[b][c][idx[k]] - pi[b][c] ----------------
__global__ __launch_bounds__(256) void gather_rel_kernel(const float* __restrict__ h,
                                                         const float* __restrict__ pi,
                                                         const int* __restrict__ idx,
                                                         float* __restrict__ rel) {
  __shared__ int sidx[KNN];
  __shared__ float spi[32];
  const int b = blockIdx.x >> 4;          // 16 c-chunks of 32
  const int c0 = (blockIdx.x & 15) * 32;
  const int k = threadIdx.x;
  sidx[k] = idx[b * KNN + k];
  if (k < 32) spi[k] = pi[b * C + c0 + k];
  __syncthreads();
  const size_t hb = (size_t)b * C * N;
  float* relb = rel + (size_t)b * C * KNN;
#pragma unroll 4
  for (int cc = 0; cc < 32; ++cc) {
    const int c = c0 + cc;
    relb[(size_t)c * KNN + k] = h[hb + (size_t)c * N + sidx[k]] - spi[cc];
  }
}

// ---------------- 4) conv GEMM: t[b][o][k] = sum_c W[o][c]*rel[b][c][k] + bias[o]
// Each wave computes a 64(M=o) x 16(N=k) tile: B fragment shared across 4 WMMAs.
__global__ __launch_bounds__(256) void conv_gemm_kernel(const float* __restrict__ W,
                                                        const float* __restrict__ bias,
                                                        const float* __restrict__ rel,
                                                        float* __restrict__ t) {
  const int b = blockIdx.x >> 4;                 // 16 blocks/batch, 8 waves each
  const int grp = blockIdx.x & 15;
  const int wave = threadIdx.x >> 5;
  const int tile = grp * 8 + wave;               // 128 wave-tiles/batch
  const int m0 = (tile >> 4) * 64;               // o  (8 M-groups)
  const int n0 = (tile & 15) * 16;               // k  (16 N-groups)
  const int lane = threadIdx.x & 31;
  const int half = lane >> 4, lm = lane & 15;
  const float* relb = rel + (size_t)b * C * KNN;
  const float* Bcol = relb + n0 + lm;
  v8f acc[4] = {v8f{}, v8f{}, v8f{}, v8f{}};
  for (int c = 0; c < C; c += 4) {
    const int kk = c + (half << 1);
    v2f bf;
    bf.x = Bcol[(size_t)kk * KNN];
    bf.y = Bcol[(size_t)(kk + 1) * KNN];
#pragma unroll
    for (int mi = 0; mi < 4; ++mi) {
      const float* ar = W + (size_t)(m0 + mi * 16 + lm) * C;   // contiguous pair -> b64
      v2f a; a.x = ar[kk]; a.y = ar[kk + 1];
      acc[mi] = __builtin_amdgcn_wmma_f32_16x16x4_f32(false, a, false, bf, (short)0,
                                                      acc[mi], false, false);
    }
  }
  float* tb = t + (size_t)b * O * KNN;
#pragma unroll
  for (int mi = 0; mi < 4; ++mi)
#pragma unroll
    for (int r = 0; r < 8; ++r) {
      const int o = m0 + mi * 16 + r + 8 * half;
      tb[(size_t)o * KNN + n0 + lm] = acc[mi][r] + bias[o];
    }
}

// ---------------- 5) Gram: S[b][k][l] = sum_o t[b][o][k] * t[b][o][l]
// 64(M=k) x 16(N=l) per wave; A[m][kk] = t[kk][m] (asM=1, asK=KNN), B[kk][n] = t[kk][n].
__global__ __launch_bounds__(256) void gram_kernel(const float* __restrict__ t,
                                                   float* __restrict__ S) {
  const int b = blockIdx.x >> 3;                 // 8 blocks/batch, 8 waves each
  const int grp = blockIdx.x & 7;
  const int wave = threadIdx.x >> 5;
  const int tile = grp * 8 + wave;               // 64 wave-tiles/batch
  const int m0 = (tile >> 4) * 64;               // k  (4 M-groups)
  const int n0 = (tile & 15) * 16;               // l  (16 N-groups)
  const int lane = threadIdx.x & 31;
  const int half = lane >> 4, lm = lane & 15;
  const float* tb = t + (size_t)b * O * KNN;
  const float* Bcol = tb + n0 + lm;
  v8f acc[4] = {v8f{}, v8f{}, v8f{}, v8f{}};
  for (int c = 0; c < O; c += 4) {
    const int kk = c + (half << 1);
    v2f bf;
    bf.x = Bcol[(size_t)kk * KNN];
    bf.y = Bcol[(size_t)(kk + 1) * KNN];
#pragma unroll
    for (int mi = 0; mi < 4; ++mi) {
      const float* ac = tb + (m0 + mi * 16 + lm);
      v2f a;
      a.x = ac[(size_t)kk * KNN];
      a.y = ac[(size_t)(kk + 1) * KNN];
      acc[mi] = __builtin_amdgcn_wmma_f32_16x16x4_f32(false, a, false, bf, (short)0,
                                                      acc[mi], false, false);
    }
  }
  float* Sb = S + (size_t)b * KNN * KNN;
#pragma unroll
  for (int mi = 0; mi < 4; ++mi)
#pragma unroll
    for (int r = 0; r < 8; ++r)
      Sb[(size_t)(m0 + mi * 16 + r + 8 * half) * KNN + n0 + lm] = acc[mi][r];
}

// ---------------- 6) softmax row-stats + column sum: w[l] = sum_k softmax(S)[k][l] ----------------
__global__ __launch_bounds__(256) void softmax_colsum_kernel(const float* __restrict__ S,
                                                             float* __restrict__ wsum) {
  __shared__ float mxs[KNN], rden[KNN];
  const int b = blockIdx.x;
  const int tid = threadIdx.x;
  const int wave = tid >> 5, lane = tid & 31;
  const float* Sb = S + (size_t)b * KNN * KNN;
  for (int j = 0; j < 32; ++j) {                 // 8 waves x 32 rows
    const int r = wave * 32 + j;
    const float* row = Sb + (size_t)r * KNN;
    float vals[8], mx = -3.402823466e38f;
#pragma unroll
    for (int i = 0; i < 8; ++i) { vals[i] = row[lane + 32 * i]; mx = fmaxf(mx, vals[i]); }
    mx = wave_max(mx);
    float s = 0.0f;
#pragma unroll
    for (int i = 0; i < 8; ++i) s += __expf(vals[i] - mx);
    s = wave_sum(s);
    if (lane == 0) { mxs[r] = mx; rden[r] = 1.0f / s; }
  }
  __syncthreads();
  const int l = tid;                              // coalesced column pass
  float acc = 0.0f;
  for (int k = 0; k < KNN; ++k)
    acc += __expf(Sb[(size_t)k * KNN + l] - mxs[k]) * rden[k];
  wsum[b * KNN + l] = acc;
}

// ---------------- 7) pooled = t @ w / K ; out = relu(lin_w @ pooled + lin_b) ----------------
__global__ __launch_bounds__(256) void pooled_final_kernel(const float* __restrict__ t,
                                                           const float* __restrict__ wsum,
                                                           const float* __restrict__ lin_w,
                                                           const float* __restrict__ lin_b,
                                                           float* __restrict__ out) {
  __shared__ float swl[KNN], spooled[O];
  const int b = blockIdx.x;
  const int tid = threadIdx.x;
  const int wave = tid >> 5, lane = tid & 31;
  swl[tid] = wsum[b * KNN + tid];
  __syncthreads();
  const float* tb = t + (size_t)b * O * KNN;
  for (int j = 0; j < 64; ++j) {                 // 8 waves x 64 o-rows
    const int o = wave * 64 + j;
    const float* row = tb + (size_t)o * KNN;
    float s = 0.0f;
#pragma unroll
    for (int i = 0; i < 8; ++i) { const int l = lane + 32 * i; s = fmaf(row[l], swl[l], s); }
    s = wave_sum(s);
    if (lane == 0) spooled[o] = s * (1.0f / (float)KNN);
  }
  __syncthreads();
  // wave-per-output-row: coalesced lin_w reads, wave32 reduction
  for (int j = wave; j < O; j += 8) {
    const float* wr = lin_w + (size_t)j * O;
    float acc = 0.0f;
#pragma unroll
    for (int i = 0; i < 16; ++i) { const int o = lane + 32 * i; acc = fmaf(wr[o], spooled[o], acc); }
    acc = wave_sum(acc);
    if (lane == 0) out[b * O + j] = fmaxf(acc + lin_b[j], 0.0f);
  }
}

// ---------------- host launch ----------------
extern "C" void kernel_launch(void* const* d_in, const int* in_sizes, int n_in,
                              void* d_out, int out_size, void* d_ws, size_t ws_size,
                              hipStream_t stream) {
  const float* h      = (const float*)d_in[0];
  const float* pi     = (const float*)d_in[1];
  const float* conv_w = (const float*)d_in[2];
  const float* conv_b = (const float*)d_in[3];
  const float* lin_w  = (const float*)d_in[4];
  const float* lin_b  = (const float*)d_in[5];
  // d_in[6] is K; fixed at 256 per setup_inputs.
  float* out = (float*)d_out;
  float* ws  = (float*)d_ws;

  float* d2  = ws + OFF_D2;
  int*   idx = (int*)(ws + OFF_IDX);
  float* rel = ws + OFF_REL;
  float* t   = ws + OFF_T;
  float* S   = ws + OFF_S;
  float* w   = ws + OFF_W;

  dist2_kernel         <<<B * 8,  256, 0, stream>>>(h, pi, d2);
  topk_kernel          <<<B,     1024, 0, stream>>>(d2, idx);
  gather_rel_kernel    <<<B * 16, 256, 0, stream>>>(h, pi, idx, rel);
  conv_gemm_kernel     <<<B * 16, 256, 0, stream>>>(conv_w, conv_b, rel, t);
  gram_kernel          <<<B * 8,  256, 0, stream>>>(t, S);
  softmax_colsum_kernel<<<B,      256, 0, stream>>>(S, w);
  pooled_final_kernel  <<<B,      256, 0, stream>>>(t, w, lin_w, lin_b, out);
}